// OneHeadedAttention_34557306863876
// MI455X (gfx1250) — compile-verified
//
#include <hip/hip_runtime.h>
#include <math.h>

// CDNA5 fp32 WMMA operand types: 16x4 A / 4x16 B = 2 VGPRs, 16x16 C/D = 8 VGPRs.
typedef float v2f __attribute__((ext_vector_type(2)));
typedef float v8f __attribute__((ext_vector_type(8)));

#define WMMA_F32(a, b, c) \
  __builtin_amdgcn_wmma_f32_16x16x4_f32(false, (a), false, (b), (short)0, (c), false, false)

#define WIN 65   // LOCAL_SIZE
#define HALO 32  // (WIN-1)/2
#define SPAN 80  // 16 query rows + 64 halo keys -> key span per 16-row tile

// ---------------------------------------------------------------------------
// Fill halo rows of extended K/V buffers with bk/bv (OOB window positions
// project to exactly the bias vector, since the padded rows are zero).
// ---------------------------------------------------------------------------
__global__ void halo_fill_kernel(float* __restrict__ Kx, float* __restrict__ Vx,
                                 const float* __restrict__ bk, const float* __restrict__ bv,
                                 int l, int d, int total) {
  int idx = blockIdx.x * blockDim.x + threadIdx.x;
  if (idx >= total) return;
  int col = idx % d;
  int r   = idx / d;        // 0 .. b*64-1
  int bb  = r / 64;
  int h   = r % 64;
  int row = (h < HALO) ? h : (l + h);  // low halo [0,32) ; high halo [l+32, l+64)
  size_t off = ((size_t)bb * (l + 64) + row) * (size_t)d + col;
  Kx[off] = bk[col];
  Vx[off] = bv[col];
}

// ---------------------------------------------------------------------------
// QKV projection: out = X @ W^T + b, one wave per 16x16 output tile.
// which = 0:Q (plain layout), 1:K / 2:V (extended layout, +32 row offset).
// ---------------------------------------------------------------------------
__global__ __launch_bounds__(32) void qkv_project_kernel(
    const float* __restrict__ x,
    const float* __restrict__ Wq, const float* __restrict__ bq,
    const float* __restrict__ Wk, const float* __restrict__ bk,
    const float* __restrict__ Wv, const float* __restrict__ bv,
    float* __restrict__ Q, float* __restrict__ Kx, float* __restrict__ Vx,
    int l, int d) {
  const int lane = threadIdx.x;
  const int m    = lane & 15;          // A: row M / B: col N within tile
  const int kg   = (lane >> 4) << 1;   // K sub-group: lanes 0-15 -> K{0,1}, 16-31 -> K{2,3}
  const int hm   = (lane >> 4) * 8;    // C/D: M = vgpr + 8*(lane/16)
  const int row0 = blockIdx.x * 16;
  const int n0   = blockIdx.y * 16;
  const int which = blockIdx.z;

  const float* W    = (which == 0) ? Wq : (which == 1) ? Wk : Wv;
  const float* bias = (which == 0) ? bq : (which == 1) ? bk : bv;

  const float* arow = x + (size_t)(row0 + m) * d;  // A[m, k] = X[row0+m, k]
  const float* brow = W + (size_t)(n0 + m) * d;    // B[k, n] = W[n0+n, k]  (X @ W^T)

  v8f acc = {};
  for (int k0 = 0; k0 < d; k0 += 4) {
    v2f a, b;
    a.x = arow[k0 + kg]; a.y = arow[k0 + kg + 1];
    b.x = brow[k0 + kg]; b.y = brow[k0 + kg + 1];
    acc = WMMA_F32(a, b, acc);
  }

  const float bc = bias[n0 + m];
  for (int g = 0; g < 8; ++g) {
    float v = acc[g] + bc;
    int r = row0 + hm + g;   // global flat row in [0, b*l)
    int c = n0 + m;
    if (which == 0) {
      Q[(size_t)r * d + c] = v;
    } else {
      int bb = r / l, rr = r - bb * l;
      float* dst = (which == 1) ? Kx : Vx;
      dst[((size_t)bb * (l + 64) + HALO + rr) * (size_t)d + c] = v;
    }
  }
}

// ---------------------------------------------------------------------------
// Banded local attention: one wave per (batch, 16-row tile).
//   S(16xSPAN) = Q_tile @ K_span^T ; softmax over each row's 65-wide band ;
//   Out(16xd)  = A_band(16xSPAN) @ V_span(SPANxd)
// ---------------------------------------------------------------------------
__global__ __launch_bounds__(32) void local_attn_kernel(
    const float* __restrict__ Q, const float* __restrict__ Kx, const float* __restrict__ Vx,
    float* __restrict__ out, int l, int d, float scale) {
  __shared__ float ldsS[16][SPAN];
  __shared__ float ldsA[16][SPAN];

  const int lane = threadIdx.x;
  const int m    = lane & 15;
  const int kg   = (lane >> 4) << 1;
  const int hm   = (lane >> 4) * 8;
  const int i0   = blockIdx.x * 16;
  const int bb   = blockIdx.y;

  const float* Qb = Q  + (size_t)bb * l * d;
  const float* Kb = Kx + (size_t)bb * (l + 64) * d;
  const float* Vb = Vx + (size_t)bb * (l + 64) * d;

  // Hoist this lane's share of the Q tile A-operand (d <= 128).
  v2f aQ[32];
  {
    const float* qrow = Qb + (size_t)(i0 + m) * d;
    for (int k0 = 0; k0 < d; k0 += 4) {
      aQ[k0 >> 2].x = qrow[k0 + kg];
      aQ[k0 >> 2].y = qrow[k0 + kg + 1];
    }
  }

  // ---- scores: S[r][n] = scale * Q[i0+r] . Kext[i0+n],  n in [0,SPAN)
  for (int nt = 0; nt < SPAN / 16; ++nt) {
    const float* krow = Kb + (size_t)(i0 + nt * 16 + m) * d;  // B[k,n] = Kext[i0+nt*16+n, k]
    v8f acc = {};
    for (int k0 = 0; k0 < d; k0 += 4) {
      v2f b;
      b.x = krow[k0 + kg];
      b.y = krow[k0 + kg + 1];
      acc = WMMA_F32(aQ[k0 >> 2], b, acc);
    }
    for (int g = 0; g < 8; ++g)
      ldsS[hm + g][nt * 16 + m] = acc[g] * scale;
  }
  __syncthreads();

  // ---- banded softmax: row r uses S[r][r .. r+64]; build zeroed A matrix
  for (int idx = lane; idx < 16 * SPAN; idx += 32)
    (&ldsA[0][0])[idx] = 0.0f;

  for (int r = 0; r < 16; ++r) {
    float mx = -3.402823466e+38f;
    for (int j = lane; j < WIN; j += 32)
      mx = fmaxf(mx, ldsS[r][r + j]);
    for (int off = 16; off > 0; off >>= 1)
      mx = fmaxf(mx, __shfl_xor(mx, off, 32));
    float s = 0.0f;
    for (int j = lane; j < WIN; j += 32)
      s += __expf(ldsS[r][r + j] - mx);
    for (int off = 16; off > 0; off >>= 1)
      s += __shfl_xor(s, off, 32);
    float inv = 1.0f / s;
    for (int j = lane; j < WIN; j += 32)
      ldsA[r][r + j] = __expf(ldsS[r][r + j] - mx) * inv;
  }
  __syncthreads();

  // ---- output: Out = A(16xSPAN) @ Vext[i0 .. i0+SPAN)(SPANxd)
  v2f aA[SPAN / 4];
  for (int k0 = 0; k0 < SPAN; k0 += 4) {
    aA[k0 >> 2].x = ldsA[m][k0 + kg];
    aA[k0 >> 2].y = ldsA[m][k0 + kg + 1];
  }
  for (int nt = 0; nt < d / 16; ++nt) {
    v8f acc = {};
    for (int k0 = 0; k0 < SPAN; k0 += 4) {
      const float* vrow = Vb + (size_t)(i0 + k0 + kg) * d + nt * 16 + m;  // B[k,n] = Vext[i0+k, col]
      v2f b;
      b.x = vrow[0];
      b.y = vrow[d];
      acc = WMMA_F32(aA[k0 >> 2], b, acc);
    }
    for (int g = 0; g < 8; ++g)
      out[((size_t)bb * l + i0 + hm + g) * (size_t)d + nt * 16 + m] = acc[g];
  }
}

// ---------------------------------------------------------------------------
extern "C" void kernel_launch(void* const* d_in, const int* in_sizes, int n_in,
                              void* d_out, int out_size, void* d_ws, size_t ws_size,
                              hipStream_t stream) {
  const float* query = (const float*)d_in[0];
  // d_in[1] (key) and d_in[2] (value) are ignored by the reference.
  const float* Wq = (const float*)d_in[3];
  const float* bq = (const float*)d_in[4];
  const float* Wk = (const float*)d_in[5];
  const float* bk = (const float*)d_in[6];
  const float* Wv = (const float*)d_in[7];
  const float* bv = (const float*)d_in[8];

  const int d    = in_sizes[4];         // 128
  const int rows = in_sizes[0] / d;     // b*l = 8192
  const int b    = 2;
  const int l    = rows / b;            // 4096

  // Workspace: Q (rows*d) | Kext (b*(l+64)*d) | Vext (b*(l+64)*d)  ~= 12.2 MB fp32
  float* Q  = (float*)d_ws;
  float* Kx = Q + (size_t)rows * d;
  float* Vx = Kx + (size_t)b * (l + 64) * d;

  {
    int total = b * 64 * d;
    halo_fill_kernel<<<(total + 255) / 256, 256, 0, stream>>>(Kx, Vx, bk, bv, l, d, total);
  }
  {
    dim3 grid(rows / 16, d / 16, 3);
    qkv_project_kernel<<<grid, 32, 0, stream>>>(query, Wq, bq, Wk, bk, Wv, bv, Q, Kx, Vx, l, d);
  }
  {
    dim3 grid(l / 16, b);
    local_attn_kernel<<<grid, 32, 0, stream>>>(Q, Kx, Vx, (float*)d_out, l, d,
                                               1.0f / sqrtf((float)d));
  }
}